// GIN_MOD_Layer_5669356830723
// MI455X (gfx1250) — compile-verified
//
#include <hip/hip_runtime.h>

typedef float v2f __attribute__((ext_vector_type(2)));
typedef float v8f __attribute__((ext_vector_type(8)));

#define N_NODES 100000
#define N_EDGES 1600000
#define DD 64
#define BN_EPS 1e-5f
#define ROWS_PER_BLOCK 128   // 8 waves x 16 rows

// LDS byte offset of a __shared__ object (for async-to-LDS asm operand)
__device__ __forceinline__ unsigned lds_offset(const void* p) {
    return (unsigned)(unsigned long long)(__attribute__((address_space(3))) const void*)p;
}

// CDNA5 async copy: global -> LDS, 16 bytes per lane, tracked by ASYNCcnt
__device__ __forceinline__ void async_load_b128(unsigned lds_addr, const void* gbase, int voff) {
    asm volatile("global_load_async_to_lds_b128 %0, %1, %2 offset:0"
                 :: "v"(lds_addr), "v"(voff), "s"(gbase) : "memory");
}

__device__ __forceinline__ void wait_asynccnt0() {
    asm volatile("s_wait_asynccnt 0x0" ::: "memory");
}

// ---------------- kernel 1: zero neigh buffer + BN stat accumulators ----------------
__global__ void k_zero(float* __restrict__ neigh, float* __restrict__ stats) {
    long i = (long)blockIdx.x * blockDim.x + threadIdx.x;
    long base = i * 4;
    if (base < (long)N_NODES * DD) {
        *(float4*)(neigh + base) = make_float4(0.f, 0.f, 0.f, 0.f);
    }
    if (i < (2 * DD) / 4) {   // 128 floats: gsum[64] + gsq[64]
        *(float4*)(stats + i * 4) = make_float4(0.f, 0.f, 0.f, 0.f);
    }
}

// ---------------- kernel 2: neigh[dst] += h[src]  (16 lanes/edge, 4 dims/lane) ------
__global__ void k_scatter(const float* __restrict__ h, const int* __restrict__ src,
                          const int* __restrict__ dst, float* __restrict__ neigh) {
    long t = (long)blockIdx.x * blockDim.x + threadIdx.x;
    long e = t >> 4;
    if (e >= N_EDGES) return;
    int c = ((int)t & 15) * 4;
    int s = src[e];
    int d = dst[e];
    float4 v = *(const float4*)(h + (long)s * DD + c);
    float* p = neigh + (long)d * DD + c;
    atomicAdd(p + 0, v.x);
    atomicAdd(p + 1, v.y);
    atomicAdd(p + 2, v.z);
    atomicAdd(p + 3, v.w);
}

// ---------------- kernel 3: x2 = relu((1+eps)h+neigh @ W1 + b1) @ W2 + b2 -----------
// Also accumulates per-column sum / sum-of-squares for BatchNorm.
__launch_bounds__(256)
__global__ void k_gin_mlp(const float* __restrict__ h, const float* __restrict__ neigh,
                          const float* __restrict__ W1, const float* __restrict__ b1,
                          const float* __restrict__ W2, const float* __restrict__ b2,
                          const float* __restrict__ eps,
                          float* __restrict__ x2, float* __restrict__ gsum,
                          float* __restrict__ gsq) {
    // W pair layout: sWp[(k/2)*128 + n*2 + (k&1)] = W[k][n]
    //   -> B fragment (rows k,k+1 at column n) is ONE aligned ds_load_b64.
    __shared__ float sW1p[DD * DD];       // 16 KB
    __shared__ float sW2p[DD * DD];       // 16 KB
    __shared__ float sB1[DD];
    __shared__ float sB2[DD];
    __shared__ float sT[8][16 * DD];      // 32 KB: async weight staging, then per-wave tiles
    __shared__ float sSum[DD];
    __shared__ float sSq[DD];

    const int tid = threadIdx.x;
    float* sRaw = &sT[0][0];              // raw W1 at [0..4095], raw W2 at [4096..8191]

    // ---- async global->LDS staging of W1/W2 (b128 per lane, ASYNCcnt tracked) ----
    {
        const unsigned t_off = lds_offset(sRaw);
        #pragma unroll
        for (int j = 0; j < 4; ++j) {
            int i = tid + j * 256;        // 0..1023 b128 chunks per matrix
            async_load_b128(t_off + i * 16,          W1, i * 16);
            async_load_b128(t_off + 16384 + i * 16,  W2, i * 16);
        }
    }
    if (tid < DD) {
        sB1[tid] = b1[tid];
        sB2[tid] = b2[tid];
        sSum[tid] = 0.f;
        sSq[tid]  = 0.f;
    }
    wait_asynccnt0();
    __syncthreads();

    // ---- repack weights into k-pair interleaved layout ----
    for (int i = tid; i < DD * DD; i += 256) {
        int k = i >> 6, n = i & 63;
        int dstidx = ((k >> 1) * DD + n) * 2 + (k & 1);
        sW1p[dstidx] = sRaw[i];
        sW2p[dstidx] = sRaw[4096 + i];
    }
    __syncthreads();                      // raw area free; sT becomes per-wave tiles

    const int wave = tid >> 5;            // 0..7
    const int lane = tid & 31;
    const int lm   = lane & 15;           // A: row m     B/C/D: column n
    const int lh   = lane >> 4;           // half-wave select
    const int row0 = blockIdx.x * ROWS_PER_BLOCK + wave * 16;
    const float epsv = 1.0f + eps[0];

    float* myT = &sT[wave][0];            // wave-private 16x64 tile

    // stage x = (1+eps)*h + neigh (coalesced b128 loads, row-clamped tail)
    #pragma unroll
    for (int j = 0; j < 8; ++j) {
        int idx = j * 128 + lane * 4;     // 0..1023
        int r = idx >> 6;
        int c = idx & 63;
        int row = row0 + r;
        if (row >= N_NODES) row = N_NODES - 1;
        float4 hv = *(const float4*)(h     + (long)row * DD + c);
        float4 nv = *(const float4*)(neigh + (long)row * DD + c);
        float4 xv;
        xv.x = epsv * hv.x + nv.x;
        xv.y = epsv * hv.y + nv.y;
        xv.z = epsv * hv.z + nv.z;
        xv.w = epsv * hv.w + nv.w;
        *(float4*)(myT + idx) = xv;
    }
    // tile is wave-local: LDS store->load ordering handled by per-wave DS ordering

    // ---- GEMM1: y1 = relu(x @ W1 + b1), one 16x64 tile per wave ----
    v8f acc[4];
    #pragma unroll
    for (int nt = 0; nt < 4; ++nt) {
        float bv = sB1[nt * 16 + lm];
        #pragma unroll
        for (int vc = 0; vc < 8; ++vc) acc[nt][vc] = bv;
    }
    #pragma unroll
    for (int kk = 0; kk < 16; ++kk) {
        const int k  = kk * 4 + lh * 2;
        const int kp = kk * 2 + lh;                          // k-pair index
        v2f a = *(const v2f*)(myT + lm * DD + k);            // A: 16x4 fragment
        #pragma unroll
        for (int nt = 0; nt < 4; ++nt) {
            v2f b = *(const v2f*)(sW1p + kp * 128 + (nt * 16 + lm) * 2);
            acc[nt] = __builtin_amdgcn_wmma_f32_16x16x4_f32(
                false, a, false, b, (short)0, acc[nt], false, false);
        }
    }

    // ReLU + spill y1 tile to LDS in row-major order (cross-lane relayout for GEMM2 A)
    #pragma unroll
    for (int nt = 0; nt < 4; ++nt) {
        #pragma unroll
        for (int vr = 0; vr < 8; ++vr) {
            int m = vr + lh * 8;
            float v = acc[nt][vr];
            v = v > 0.f ? v : 0.f;
            myT[m * DD + nt * 16 + lm] = v;
        }
    }

    // ---- GEMM2: x2 = y1 @ W2 + b2 ----
    v8f acc2[4];
    #pragma unroll
    for (int nt = 0; nt < 4; ++nt) {
        float bv = sB2[nt * 16 + lm];
        #pragma unroll
        for (int vc = 0; vc < 8; ++vc) acc2[nt][vc] = bv;
    }
    #pragma unroll
    for (int kk = 0; kk < 16; ++kk) {
        const int k  = kk * 4 + lh * 2;
        const int kp = kk * 2 + lh;
        v2f a = *(const v2f*)(myT + lm * DD + k);
        #pragma unroll
        for (int nt = 0; nt < 4; ++nt) {
            v2f b = *(const v2f*)(sW2p + kp * 128 + (nt * 16 + lm) * 2);
            acc2[nt] = __builtin_amdgcn_wmma_f32_16x16x4_f32(
                false, a, false, b, (short)0, acc2[nt], false, false);
        }
    }

    // store pre-BN activations + per-column partial stats (skip clamped tail rows)
    #pragma unroll
    for (int nt = 0; nt < 4; ++nt) {
        float s = 0.f, q = 0.f;
        #pragma unroll
        for (int vr = 0; vr < 8; ++vr) {
            int m = vr + lh * 8;
            int row = row0 + m;
            float v = acc2[nt][vr];
            if (row < N_NODES) {
                x2[(long)row * DD + nt * 16 + lm] = v;
                s += v;
                q += v * v;
            }
        }
        atomicAdd(&sSum[nt * 16 + lm], s);   // ds_add_f32
        atomicAdd(&sSq[nt * 16 + lm], q);
    }
    __syncthreads();
    if (tid < DD) {                          // one global atomic per column per WG
        atomicAdd(&gsum[tid], sSum[tid]);
        atomicAdd(&gsq[tid],  sSq[tid]);
    }
}

// ---------------- kernel 4: finalize BN stats into scale/shift ----------------------
__global__ void k_bn_finalize(const float* __restrict__ gsum, const float* __restrict__ gsq,
                              const float* __restrict__ gamma, const float* __restrict__ beta,
                              float* __restrict__ scale, float* __restrict__ shift) {
    int n = threadIdx.x;
    if (n < DD) {
        float inv_n = 1.0f / (float)N_NODES;
        float mu  = gsum[n] * inv_n;
        float var = gsq[n] * inv_n - mu * mu;
        float sc  = gamma[n] * rsqrtf(var + BN_EPS);
        scale[n] = sc;
        shift[n] = beta[n] - mu * sc;
    }
}

// ---------------- kernel 5: out = h + relu(x2*scale + shift) ------------------------
__global__ void k_final(const float* __restrict__ h, const float* __restrict__ x2,
                        const float* __restrict__ scale, const float* __restrict__ shift,
                        float* __restrict__ out) {
    long i = ((long)blockIdx.x * blockDim.x + threadIdx.x) * 4;
    if (i >= (long)N_NODES * DD) return;
    int n = (int)(i & (DD - 1));             // column (rows are 64-float aligned)
    float4 xv = *(const float4*)(x2 + i);
    float4 hv = *(const float4*)(h + i);
    float4 o;
    o.x = hv.x + fmaxf(xv.x * scale[n + 0] + shift[n + 0], 0.f);
    o.y = hv.y + fmaxf(xv.y * scale[n + 1] + shift[n + 1], 0.f);
    o.z = hv.z + fmaxf(xv.z * scale[n + 2] + shift[n + 2], 0.f);
    o.w = hv.w + fmaxf(xv.w * scale[n + 3] + shift[n + 3], 0.f);
    *(float4*)(out + i) = o;
}

extern "C" void kernel_launch(void* const* d_in, const int* in_sizes, int n_in,
                              void* d_out, int out_size, void* d_ws, size_t ws_size,
                              hipStream_t stream) {
    const float* h     = (const float*)d_in[0];
    const int*   src   = (const int*)  d_in[1];
    const int*   dst   = (const int*)  d_in[2];
    const float* W1    = (const float*)d_in[3];
    const float* b1    = (const float*)d_in[4];
    const float* W2    = (const float*)d_in[5];
    const float* b2    = (const float*)d_in[6];
    const float* gamma = (const float*)d_in[7];
    const float* beta  = (const float*)d_in[8];
    const float* eps   = (const float*)d_in[9];
    float* out = (float*)d_out;

    // workspace layout (floats): neigh[N*D] | x2[N*D] | gsum[64] | gsq[64] | scale[64] | shift[64]
    float* ws    = (float*)d_ws;
    float* neigh = ws;
    float* x2    = ws + (long)N_NODES * DD;
    float* gsum  = ws + 2L * N_NODES * DD;
    float* gsq   = gsum + DD;
    float* scale = gsq + DD;
    float* shift = scale + DD;

    const long nd = (long)N_NODES * DD;                       // 6,400,000
    const int zero_blocks    = (int)((nd / 4 + 255) / 256);   // 6250
    const int scatter_blocks = (int)(((long)N_EDGES * 16 + 255) / 256);
    const int mlp_blocks     = (N_NODES + ROWS_PER_BLOCK - 1) / ROWS_PER_BLOCK;  // 782

    k_zero<<<zero_blocks, 256, 0, stream>>>(neigh, gsum);
    k_scatter<<<scatter_blocks, 256, 0, stream>>>(h, src, dst, neigh);
    k_gin_mlp<<<mlp_blocks, 256, 0, stream>>>(h, neigh, W1, b1, W2, b2, eps, x2, gsum, gsq);
    k_bn_finalize<<<1, 64, 0, stream>>>(gsum, gsq, gamma, beta, scale, shift);
    k_final<<<zero_blocks, 256, 0, stream>>>(h, x2, scale, shift, out);
}